// GNN_2156073582596
// MI455X (gfx1250) — compile-verified
//
#include <hip/hip_runtime.h>
#include <math.h>

#define N_NODES   100000
#define N_EDGES   1600000
#define N_FEAT    128
#define HIDDEN    64
#define N_CLASSES 15
#define N_GRAPHS  256

typedef __attribute__((ext_vector_type(2))) float v2f;
typedef __attribute__((ext_vector_type(8))) float v8f;

// Fire-and-forget f32 atomic add -> global_atomic_add_f32 (CDNA5).
// Tracked by STOREcnt; S_ENDPGM implicitly waits, and stream ordering makes
// results visible to the next kernel.
__device__ __forceinline__ void gatomic_add_f32(float* p, float v) {
    asm volatile("global_atomic_add_f32 %0, %1, off" : : "v"(p), "v"(v) : "memory");
}

// ---------------- degree / norm ----------------
__global__ void k_fill1(float* deg, int n) {
    int t = blockIdx.x * 256 + threadIdx.x;
    if (t < n) deg[t] = 1.0f;          // self-loop contributes 1 to every degree
}

__global__ void k_deg_edges(const int* __restrict__ dst, float* deg) {
    int t = blockIdx.x * 256 + threadIdx.x;
    if (t < N_EDGES) gatomic_add_f32(deg + dst[t], 1.0f);
}

__global__ void k_rsqrt(float* dinv, int n) {
    int t = blockIdx.x * 256 + threadIdx.x;
    if (t < n) {
        float d = dinv[t];
        dinv[t] = (d > 0.0f) ? rsqrtf(d) : 0.0f;
    }
}

// ---------------- fp32 WMMA GEMM: out[M x 64] = A[M x K] @ W[K x 64] -------
// One wave -> 16x64 tile via 4 accumulators of V_WMMA_F32_16X16X4_F32.
// W is staged in LDS pre-packed in B-fragment order so each B operand is a
// single aligned ds_load_b64:
//   sW[(k>>1)*128 + col*2 + (k&1)] = W[k*64 + col]
// B fragment for k-step k0, half-wave hi, column col: pair p = k0/2 + hi,
//   { W[(k0+2*hi)*64+col], W[(k0+2*hi+1)*64+col] } = (v2f)&sW[p*128 + col*2]
__global__ __launch_bounds__(256) void k_gemm_wmma(const float* __restrict__ A,
                                                   const float* __restrict__ W,
                                                   float* __restrict__ out,
                                                   int M, int K) {
    __shared__ float sW[N_FEAT * HIDDEN];      // up to 128x64 = 32 KB
    const int tid = threadIdx.x;
    const int kw = K * HIDDEN;
    for (int i = tid; i < kw; i += 256) {
        int k = i >> 6, col = i & 63;
        sW[(k >> 1) * 128 + col * 2 + (k & 1)] = W[i];
    }
    __syncthreads();

    const int lane = tid & 31;
    const int wave = tid >> 5;
    const int row0 = (blockIdx.x * 8 + wave) * 16;
    if (row0 >= M) return;

    const int lo = lane & 15;                  // column / M-row within half-wave
    const int hi = lane >> 4;                  // half-wave select

    v8f acc0 = {}, acc1 = {}, acc2 = {}, acc3 = {};
    // A fragment (16x4 fp32): lane<16 holds K={k,k+1}; lane>=16 holds K={k+2,k+3}
    const float* arow = A + (long long)(row0 + lo) * K + 2 * hi;

    v2f a = *(const v2f*)(arow);               // software-pipelined A fragment
    for (int k = 0; k < K; k += 4) {
        v2f acur = a;
        if (k + 4 < K) a = *(const v2f*)(arow + k + 4);
        const v2f* wb = (const v2f*)(sW + ((k >> 1) + hi) * 128 + lo * 2);
        v2f b0 = wb[0];
        v2f b1 = wb[16];
        v2f b2 = wb[32];
        v2f b3 = wb[48];
        acc0 = __builtin_amdgcn_wmma_f32_16x16x4_f32(false, acur, false, b0, (short)0, acc0, false, false);
        acc1 = __builtin_amdgcn_wmma_f32_16x16x4_f32(false, acur, false, b1, (short)0, acc1, false, false);
        acc2 = __builtin_amdgcn_wmma_f32_16x16x4_f32(false, acur, false, b2, (short)0, acc2, false, false);
        acc3 = __builtin_amdgcn_wmma_f32_16x16x4_f32(false, acur, false, b3, (short)0, acc3, false, false);
    }

    // C/D layout: VGPR v -> M = row0 + v + 8*hi, N = n*16 + lo
    float* orow = out + (long long)(row0 + 8 * hi) * HIDDEN + lo;
#pragma unroll
    for (int v = 0; v < 8; ++v) {
        orow[v * HIDDEN +  0] = acc0[v];
        orow[v * HIDDEN + 16] = acc1[v];
        orow[v * HIDDEN + 32] = acc2[v];
        orow[v * HIDDEN + 48] = acc3[v];
    }
}

// ---------------- bias + self-loop init: out = b + dinv^2 * xw -------------
__global__ void k_self_bias(const float* __restrict__ xw,
                            const float* __restrict__ dinv,
                            const float* __restrict__ b,
                            float* __restrict__ out) {
    int t = blockIdx.x * 256 + threadIdx.x;
    if (t >= N_NODES * HIDDEN) return;
    int i = t >> 6, f = t & 63;
    float dv = dinv[i];
    out[t] = b[f] + dv * dv * xw[t];
}

// ---------------- edge scatter: out[dst] += dinv[s]*dinv[d] * xw[src] ------
// 64 threads per edge; e = t>>6 is wave-uniform (aligned 64-thread groups),
// so scalarize the indices -> SMEM loads + SADDR-form atomics.
__global__ void k_scatter(const int* __restrict__ src, const int* __restrict__ dst,
                          const float* __restrict__ dinv,
                          const float* __restrict__ xw,
                          float* __restrict__ out) {
    int t = blockIdx.x * 256 + threadIdx.x;   // N_EDGES*64 = 102.4M < 2^31
    if (t >= N_EDGES * HIDDEN) return;
    int e = t >> 6, f = t & 63;
    int s = __builtin_amdgcn_readfirstlane(src[e]);
    int d = __builtin_amdgcn_readfirstlane(dst[e]);
    float nrm = dinv[s] * dinv[d];
    gatomic_add_f32(out + d * HIDDEN + f, nrm * xw[s * HIDDEN + f]);
}

__global__ void k_relu(float* h, int n) {
    int t = blockIdx.x * 256 + threadIdx.x;
    if (t < n) h[t] = fmaxf(h[t], 0.0f);
}

// ---------------- pooling --------------------------------------------------
__global__ void k_zero(float* p, int n) {
    int t = blockIdx.x * 256 + threadIdx.x;
    if (t < n) p[t] = 0.0f;
}

__global__ void k_pool(const float* __restrict__ h, const int* __restrict__ batch,
                       float* __restrict__ pooled, float* __restrict__ cnts) {
    int t = blockIdx.x * 256 + threadIdx.x;
    if (t >= N_NODES * HIDDEN) return;
    int i = t >> 6, f = t & 63;
    int g = __builtin_amdgcn_readfirstlane(batch[i]);   // i is wave-uniform
    gatomic_add_f32(pooled + g * HIDDEN + f, h[t]);
    if (f == 0) gatomic_add_f32(cnts + g, 1.0f);
}

// ---------------- FC + log_softmax ----------------------------------------
__global__ void k_head(const float* __restrict__ pooled, const float* __restrict__ cnts,
                       const float* __restrict__ fcw, const float* __restrict__ fcb,
                       float* __restrict__ out) {
    int g = threadIdx.x;                       // 256 graphs, one block
    if (g >= N_GRAPHS) return;
    float inv = 1.0f / fmaxf(cnts[g], 1.0f);
    float logits[N_CLASSES];
#pragma unroll
    for (int c = 0; c < N_CLASSES; ++c) logits[c] = fcb[c];
    for (int k = 0; k < HIDDEN; ++k) {
        float p = pooled[g * HIDDEN + k] * inv;
#pragma unroll
        for (int c = 0; c < N_CLASSES; ++c) logits[c] += p * fcw[k * N_CLASSES + c];
    }
    float m = logits[0];
#pragma unroll
    for (int c = 1; c < N_CLASSES; ++c) m = fmaxf(m, logits[c]);
    float s = 0.0f;
#pragma unroll
    for (int c = 0; c < N_CLASSES; ++c) s += expf(logits[c] - m);
    float lse = logf(s);
#pragma unroll
    for (int c = 0; c < N_CLASSES; ++c) out[g * N_CLASSES + c] = logits[c] - m - lse;
}

extern "C" void kernel_launch(void* const* d_in, const int* in_sizes, int n_in,
                              void* d_out, int out_size, void* d_ws, size_t ws_size,
                              hipStream_t stream) {
    const float* x    = (const float*)d_in[0];
    const float* w1   = (const float*)d_in[1];
    const float* b1   = (const float*)d_in[2];
    const float* w2   = (const float*)d_in[3];
    const float* b2   = (const float*)d_in[4];
    const float* w3   = (const float*)d_in[5];
    const float* b3   = (const float*)d_in[6];
    const float* fcw  = (const float*)d_in[7];
    const float* fcb  = (const float*)d_in[8];
    const int*   eidx = (const int*)d_in[9];       // [2, E]: src row then dst row
    const int*   batch= (const int*)d_in[10];
    float* out = (float*)d_out;

    const int* src = eidx;
    const int* dst = eidx + N_EDGES;

    float* ws     = (float*)d_ws;
    float* xw     = ws;                               // 100000*64
    float* bufA   = xw   + N_NODES * HIDDEN;          // 100000*64
    float* bufB   = bufA + N_NODES * HIDDEN;          // 100000*64
    float* dinv   = bufB + N_NODES * HIDDEN;          // 100000 (deg, then d^-1/2)
    float* pooled = dinv + N_NODES;                   // 256*64
    float* cnts   = pooled + N_GRAPHS * HIDDEN;       // 256

    const int nBlkNode   = (N_NODES + 255) / 256;
    const int nBlkEdge   = (N_EDGES + 255) / 256;
    const int nBlkNF     = (N_NODES * HIDDEN + 255) / 256;   // 25000
    const int nBlkEF     = (N_EDGES * HIDDEN + 255) / 256;   // 400000
    const int nBlkGemm   = (N_NODES + 127) / 128;            // 8 waves x 16 rows

    // degrees -> dinv
    k_fill1   <<<nBlkNode, 256, 0, stream>>>(dinv, N_NODES);
    k_deg_edges<<<nBlkEdge, 256, 0, stream>>>(dst, dinv);
    k_rsqrt   <<<nBlkNode, 256, 0, stream>>>(dinv, N_NODES);

    // layer 1: x[N,128] @ w1 -> xw; bufA = conv; relu
    k_gemm_wmma<<<nBlkGemm, 256, 0, stream>>>(x, w1, xw, N_NODES, N_FEAT);
    k_self_bias<<<nBlkNF, 256, 0, stream>>>(xw, dinv, b1, bufA);
    k_scatter  <<<nBlkEF, 256, 0, stream>>>(src, dst, dinv, xw, bufA);
    k_relu     <<<nBlkNF, 256, 0, stream>>>(bufA, N_NODES * HIDDEN);

    // layer 2
    k_gemm_wmma<<<nBlkGemm, 256, 0, stream>>>(bufA, w2, xw, N_NODES, HIDDEN);
    k_self_bias<<<nBlkNF, 256, 0, stream>>>(xw, dinv, b2, bufB);
    k_scatter  <<<nBlkEF, 256, 0, stream>>>(src, dst, dinv, xw, bufB);
    k_relu     <<<nBlkNF, 256, 0, stream>>>(bufB, N_NODES * HIDDEN);

    // layer 3
    k_gemm_wmma<<<nBlkGemm, 256, 0, stream>>>(bufB, w3, xw, N_NODES, HIDDEN);
    k_self_bias<<<nBlkNF, 256, 0, stream>>>(xw, dinv, b3, bufA);
    k_scatter  <<<nBlkEF, 256, 0, stream>>>(src, dst, dinv, xw, bufA);
    k_relu     <<<nBlkNF, 256, 0, stream>>>(bufA, N_NODES * HIDDEN);

    // mean pool + head
    k_zero<<<((N_GRAPHS * HIDDEN + N_GRAPHS) + 255) / 256, 256, 0, stream>>>(
        pooled, N_GRAPHS * HIDDEN + N_GRAPHS);
    k_pool<<<nBlkNF, 256, 0, stream>>>(bufA, batch, pooled, cnts);
    k_head<<<1, 256, 0, stream>>>(pooled, cnts, fcw, fcb, out);
}